// NeighboursToNodesCollector_65249143161004
// MI455X (gfx1250) — compile-verified
//
#include <hip/hip_runtime.h>
#include <hip/hip_bf16.h>
#include <stdint.h>

// Problem constants from the reference: F=32 floats/row (128B), MAX_DEG=4.
#define FEAT 32
#define MAXD 4

// ---- CDNA5 async global<->LDS DMA path (probe via __has_builtin) ----------
#if defined(__HIP_DEVICE_COMPILE__) && defined(__gfx1250__) && \
    __has_builtin(__builtin_amdgcn_global_load_async_to_lds_b128) && \
    __has_builtin(__builtin_amdgcn_global_store_async_from_lds_b128)
#define USE_ASYNC_LDS 1
#else
#define USE_ASYNC_LDS 0
#endif

// The builtins want pointers to a 4 x i32 vector in the right address space
// (per the hipcc diagnostic: "__attribute__((__vector_size__(4*sizeof(int)))) int *").
typedef int v4i __attribute__((ext_vector_type(4)));
typedef __attribute__((address_space(1))) v4i gv4i; // global
typedef __attribute__((address_space(3))) v4i lv4i; // LDS

__device__ __forceinline__ void wait_async0() {
#if defined(__HIP_DEVICE_COMPILE__) && defined(__gfx1250__) && \
    __has_builtin(__builtin_amdgcn_s_wait_asynccnt)
  __builtin_amdgcn_s_wait_asynccnt(0);
#else
  asm volatile("s_wait_asynccnt 0" ::: "memory");
#endif
}

// LDS staging: 256 lanes * 144B (128B row + 16B pad to spread banks)
#define STAGE_STRIDE_F 36

// Copy one 128-byte feature row src(global) -> dst(global).
// Async path: DMA global->LDS, wait asynccnt, DMA LDS->global (no VGPR data).
__device__ __forceinline__ void copy_row128(const float* __restrict__ src,
                                            float* __restrict__ dst,
                                            float* lds_stage) {
#if USE_ASYNC_LDS
  gv4i* gs = (gv4i*)src;
  gv4i* gd = (gv4i*)dst;
  lv4i* l  = (lv4i*)lds_stage;
#define ALOAD(off)  __builtin_amdgcn_global_load_async_to_lds_b128(gs, l, off, 0)
#define ASTORE(off) __builtin_amdgcn_global_store_async_from_lds_b128(gd, l, off, 0)
  ALOAD(0);  ALOAD(16); ALOAD(32); ALOAD(48);
  ALOAD(64); ALOAD(80); ALOAD(96); ALOAD(112);
  wait_async0();
  ASTORE(0);  ASTORE(16); ASTORE(32); ASTORE(48);
  ASTORE(64); ASTORE(80); ASTORE(96); ASTORE(112);
  wait_async0();
#undef ALOAD
#undef ASTORE
#else
  (void)lds_stage;
  const float4* s4 = (const float4*)src;
  float4* d4 = (float4*)dst;
#pragma unroll
  for (int j = 0; j < 8; ++j) d4[j] = s4[j];
#endif
}

// Kernel 1: zero the degree counters (2*N ints in workspace).
__global__ void k_zero_i32(int* __restrict__ p, int n) {
  int i = blockIdx.x * blockDim.x + threadIdx.x;
  if (i < n) p[i] = 0;
}

// Kernel 2: sender-block scatter. slot = rank among edges with senders==s.
__global__ void k_scatter_send(const float* __restrict__ nodes,
                               const int* __restrict__ senders,
                               const int* __restrict__ receivers,
                               int* __restrict__ deg_s,
                               float* __restrict__ out, int E) {
#if USE_ASYNC_LDS
  __shared__ float stage[256 * STAGE_STRIDE_F];
  float* my_stage = stage + threadIdx.x * STAGE_STRIDE_F;
#else
  float* my_stage = nullptr;
#endif
  int e = blockIdx.x * blockDim.x + threadIdx.x;
  if (e >= E) return;
  int s = senders[e];
  int r = receivers[e];
  int slot = atomicAdd(&deg_s[s], 1);
  if (slot < MAXD) {
    const float* src = nodes + (size_t)r * FEAT;
    float* dst = out + (size_t)s * (MAXD * FEAT) + (size_t)slot * FEAT;
    copy_row128(src, dst, my_stage);
  }
}

// Kernel 3: receiver-block scatter. slot = deg_s[r] (final) + rank in recv block.
__global__ void k_scatter_recv(const float* __restrict__ nodes,
                               const int* __restrict__ senders,
                               const int* __restrict__ receivers,
                               const int* __restrict__ deg_s,
                               int* __restrict__ deg_r,
                               float* __restrict__ out, int E) {
#if USE_ASYNC_LDS
  __shared__ float stage[256 * STAGE_STRIDE_F];
  float* my_stage = stage + threadIdx.x * STAGE_STRIDE_F;
#else
  float* my_stage = nullptr;
#endif
  int e = blockIdx.x * blockDim.x + threadIdx.x;
  if (e >= E) return;
  int s = senders[e];
  int r = receivers[e];
  int slot = deg_s[r] + atomicAdd(&deg_r[r], 1);
  if (slot < MAXD) {
    const float* src = nodes + (size_t)s * FEAT;
    float* dst = out + (size_t)r * (MAXD * FEAT) + (size_t)slot * FEAT;
    copy_row128(src, dst, my_stage);
  }
}

// Kernel 4: zero the pad slots (slot >= total degree). One thread per 16B chunk:
// chunk id t -> node = t/32, slot = (t%32)/8. Fully coalesced b128 stores.
__global__ void k_zero_pad(const int* __restrict__ deg_s,
                           const int* __restrict__ deg_r,
                           float4* __restrict__ out4, long long nchunks) {
  long long t = (long long)blockIdx.x * blockDim.x + threadIdx.x;
  if (t >= nchunks) return;
  int node = (int)(t >> 5);
  int slot = (int)((t >> 3) & 3);
  int d = deg_s[node] + deg_r[node];
  if (slot >= d) {
    float4 z;
    z.x = 0.f; z.y = 0.f; z.z = 0.f; z.w = 0.f;
    out4[t] = z;
  }
}

extern "C" void kernel_launch(void* const* d_in, const int* in_sizes, int n_in,
                              void* d_out, int out_size, void* d_ws, size_t ws_size,
                              hipStream_t stream) {
  (void)n_in; (void)out_size; (void)ws_size;
  const float* nodes   = (const float*)d_in[0];
  // d_in[1] = edges, unused by the forward math
  const int* senders   = (const int*)d_in[2];
  const int* receivers = (const int*)d_in[3];
  float* out = (float*)d_out;

  const int N = in_sizes[0] / FEAT;   // 1,000,000
  const int E = in_sizes[2];          // 1,000,000

  int* deg_s = (int*)d_ws;            // [N]
  int* deg_r = deg_s + N;             // [N]

  const int TB = 256;

  // 1) zero degree counters (must happen every call: counters are reused)
  {
    int n = 2 * N;
    k_zero_i32<<<(n + TB - 1) / TB, TB, 0, stream>>>(deg_s, n);
  }
  // 2) sender-block scatter (fills slots [0, deg_s))
  k_scatter_send<<<(E + TB - 1) / TB, TB, 0, stream>>>(
      nodes, senders, receivers, deg_s, out, E);
  // 3) receiver-block scatter (fills slots [deg_s, deg_s+deg_r)); separate
  //    launch guarantees deg_s is final before it is read
  k_scatter_recv<<<(E + TB - 1) / TB, TB, 0, stream>>>(
      nodes, senders, receivers, deg_s, deg_r, out, E);
  // 4) zero the remaining pad slots
  {
    long long nchunks = (long long)N * (MAXD * FEAT / 4); // 16B chunks
    long long nblocks = (nchunks + TB - 1) / TB;
    k_zero_pad<<<(unsigned)nblocks, TB, 0, stream>>>(deg_s, deg_r,
                                                     (float4*)out, nchunks);
  }
}